// ModifyTrend1_15513421873614
// MI455X (gfx1250) — compile-verified
//
#include <hip/hip_runtime.h>

// ---------------- problem constants ----------------
#define DAYS   365
#define NYEAR  30
#define NY     3
#define NT     (NYEAR * DAYS)       // 10950
#define NGAGE  2000
#define NPAIR  435                  // 30*29/2 year pairs
#define NK     5                    // percentiles
#define NTASK  (2 * NK * NGAGE)     // 20000 (tensor,k,gage) columns
#define QSTRIDE 32                  // padded year stride in Q buffer
#define MED_RANK 217                // (435-1)//2 lower median
#define SORTN  512                  // bitonic size (365 padded with +INF)

typedef __attribute__((ext_vector_type(2))) float v2f;
typedef __attribute__((ext_vector_type(8))) float v8f;

#if defined(__has_builtin)
#if __has_builtin(__builtin_amdgcn_wmma_f32_16x16x4_f32)
#define HAVE_WMMA_F32X4 1
#endif
#endif

// =====================================================================
// Kernel 1: one wave32 per (tensor, gage, year). Bitonic-sort 365 values
// (padded to 512 with +INF) in a private LDS region, then emit the 5
// order statistics. Full sort => tie handling is free (multiset ranks).
// 45 passes x 8 compare-exchanges/lane; bitonic index pattern hits 32
// distinct LDS banks per access (conflict-free with 64 banks / 32 lanes).
// =====================================================================
__global__ __launch_bounds__(256) void quantile_kernel(
    const float* __restrict__ outp, const float* __restrict__ targ,
    float* __restrict__ Qbuf) {
  const int lane = threadIdx.x & 31;
  const int wave = threadIdx.x >> 5;
  const int task = blockIdx.x * 8 + wave;          // exact grid: 120000 waves

  const int tensor = task / (NGAGE * NYEAR);
  const int rest   = task % (NGAGE * NYEAR);
  const int g      = rest / NYEAR;
  const int y      = rest % NYEAR;

  // only ny==0 slice is used: element [g][0][y*365+d]
  const float* src = (tensor == 0 ? outp : targ)
                   + (size_t)g * (NY * NT) + (size_t)y * DAYS;

  __shared__ float sbuf[8][SORTN];
  float* a = sbuf[wave];

  const float INF = __builtin_inff();
#pragma unroll
  for (int r = 0; r < SORTN / 32; ++r) {           // 16 iterations
    const int e = lane + 32 * r;
    a[e] = (e < DAYS) ? src[e] : INF;              // coalesced b32 loads
  }
  __builtin_amdgcn_wave_barrier();

  // ascending bitonic sort of 512 elements, wave-private region
  for (int kk = 2; kk <= SORTN; kk <<= 1) {
    for (int j = kk >> 1; j > 0; j >>= 1) {
#pragma unroll
      for (int t = 0; t < SORTN / 64; ++t) {       // 8 pairs per lane
        const int p  = lane + 32 * t;              // pair id 0..255
        const int i  = ((p & ~(j - 1)) << 1) | (p & (j - 1)); // bit j clear
        const int ix = i | j;                      // partner
        const bool up = ((i & kk) == 0);
        const float va = a[i];
        const float vb = a[ix];
        const float lo = fminf(va, vb);
        const float hi = fmaxf(va, vb);
        a[i]  = up ? lo : hi;
        a[ix] = up ? hi : lo;
      }
      __builtin_amdgcn_wave_barrier();             // scheduling fence only
    }
  }

  // kidx = [364, 357, 182, 109, 7] for PERCENT_LST = [100, 98, 50, 30, 2]
  if (lane < NK) {
    int kidx;
    switch (lane) {
      case 0: kidx = 364; break;
      case 1: kidx = 357; break;
      case 2: kidx = 182; break;
      case 3: kidx = 109; break;
      default: kidx = 7;  break;
    }
    const int t = (tensor * NK + lane) * NGAGE + g;
    Qbuf[(size_t)t * QSTRIDE + y] = a[kidx];
  }
}

// =====================================================================
// Kernel 2: per block of 16 task-columns:
//   S(448x16) = A(448x32) x Q(32x16) via V_WMMA_F32_16X16X4_F32,
//   A generated procedurally (sparse pair-difference matrix),
//   then lower-median (rank 217 of 435) per column via counting rank.
// =====================================================================
__global__ __launch_bounds__(256) void trend_kernel(
    const float* __restrict__ Qbuf, float* __restrict__ mQ) {
  const int lane  = threadIdx.x & 31;
  const int wave  = threadIdx.x >> 5;   // 0..7
  const int task0 = blockIdx.x * 16;

  __shared__ float Qs[32][17];          // [year(k) padded to 32][col], pitch 17
  __shared__ float S[448][17];          // slope tile, pitch 17 breaks bank conflicts

  // stage Q columns into LDS (zero-pad years 30,31)
  for (int idx = threadIdx.x; idx < 32 * 16; idx += blockDim.x) {
    const int k = idx >> 4, col = idx & 15;
    Qs[k][col] = (k < NYEAR) ? Qbuf[(size_t)(task0 + col) * QSTRIDE + k] : 0.0f;
  }
  __syncthreads();

#if defined(HAVE_WMMA_F32X4)
  // ---- WMMA path: 28 row tiles of 16, K = 32 in 8 steps of 4 ----
  for (int rt = wave; rt < 28; rt += 8) {
    const int m   = lane & 15;          // A: lanes 0-15 & 16-31 both hold M=0..15
    const int row = rt * 16 + m;

    // decode pair (pi, pj) for this slope row; A[row][k] = inv*(k==pj) - inv*(k==pi)
    int pi = -2, pj = -1;
    float inv = 0.0f;
    if (row < NPAIR) {
      int i = 0, rem = row;
      while (rem >= (NYEAR - 1) - i) { rem -= (NYEAR - 1) - i; ++i; }
      pi = i; pj = i + 1 + rem;
      inv = 1.0f / (float)(pj - pi);
    }

    const int kb = (lane < 16) ? 0 : 2;  // lane halves carry K pairs (0,1)/(2,3)
    v8f acc = {0.f, 0.f, 0.f, 0.f, 0.f, 0.f, 0.f, 0.f};
#pragma unroll
    for (int kk = 0; kk < 8; ++kk) {
      const int k0 = kk * 4 + kb;
      v2f a, b;
      a.x = (k0     == pj) ? inv : ((k0     == pi) ? -inv : 0.0f);
      a.y = (k0 + 1 == pj) ? inv : ((k0 + 1 == pi) ? -inv : 0.0f);
      b.x = Qs[k0][m];
      b.y = Qs[k0 + 1][m];
      acc = __builtin_amdgcn_wmma_f32_16x16x4_f32(
          /*neg_a=*/false, a, /*neg_b=*/false, b,
          /*c_mod=*/(short)0, acc, /*reuse_a=*/false, /*reuse_b=*/false);
    }
    // C/D layout: VGPR r -> M=r (lanes 0-15) / M=r+8 (lanes 16-31), N=lane&15
    const int roff = (lane < 16) ? 0 : 8;
#pragma unroll
    for (int r = 0; r < 8; ++r)
      S[rt * 16 + r + roff][m] = acc[r];
  }
#else
  // ---- fallback: direct sparse slope computation ----
  for (int row = threadIdx.x; row < 448; row += blockDim.x) {
    if (row < NPAIR) {
      int i = 0, rem = row;
      while (rem >= (NYEAR - 1) - i) { rem -= (NYEAR - 1) - i; ++i; }
      const int pj = i + 1 + rem;
      const float inv = 1.0f / (float)(pj - i);
      for (int col = 0; col < 16; ++col)
        S[row][col] = (Qs[pj][col] - Qs[i][col]) * inv;
    }
  }
#endif
  __syncthreads();

  // ---- lower median per column: counting rank, 2 columns per wave ----
  for (int cc = 0; cc < 2; ++cc) {
    const int col = wave + cc * 8;
    float x[14];
    int   cnt[14];
#pragma unroll
    for (int r = 0; r < 14; ++r) {
      const int e = lane + 32 * r;
      x[r] = (e < NPAIR) ? S[e][col] : 0.0f;
      cnt[r] = 0;
    }
    for (int j = 0; j < NPAIR; ++j) {
      const float v = S[j][col];  // broadcast read
#pragma unroll
      for (int r = 0; r < 14; ++r) {
        const int e = lane + 32 * r;
        cnt[r] += (v < x[r] || (v == x[r] && j < e)) ? 1 : 0;
      }
    }
#pragma unroll
    for (int r = 0; r < 14; ++r) {
      const int e = lane + 32 * r;
      if (e < NPAIR && cnt[r] == MED_RANK) mQ[task0 + col] = x[r];
    }
  }
}

// =====================================================================
// Kernel 3: deterministic single-block reduction
//   out = (1/NGAGE) * sum_{k,g} (mp[k][g] - mt[k][g])^2
// =====================================================================
__global__ __launch_bounds__(256) void reduce_kernel(
    const float* __restrict__ mQ, float* __restrict__ out) {
  __shared__ float red[256];
  float s = 0.0f;
  for (int i = threadIdx.x; i < NK * NGAGE; i += 256) {
    const float d = mQ[i] - mQ[i + NK * NGAGE];
    s += d * d;
  }
  red[threadIdx.x] = s;
  __syncthreads();
  for (int off = 128; off > 0; off >>= 1) {
    if ((int)threadIdx.x < off) red[threadIdx.x] += red[threadIdx.x + off];
    __syncthreads();
  }
  if (threadIdx.x == 0) out[0] = red[0] * (1.0f / (float)NGAGE);
}

extern "C" void kernel_launch(void* const* d_in, const int* in_sizes, int n_in,
                              void* d_out, int out_size, void* d_ws, size_t ws_size,
                              hipStream_t stream) {
  const float* outp = (const float*)d_in[0];
  const float* targ = (const float*)d_in[1];

  float* Qbuf = (float*)d_ws;                        // NTASK*QSTRIDE floats (2.56 MB)
  float* mQ   = Qbuf + (size_t)NTASK * QSTRIDE;      // NTASK floats

  // 1) 120000 wave-tasks, 8 waves/block
  quantile_kernel<<<(2 * NGAGE * NYEAR) / 8, 256, 0, stream>>>(outp, targ, Qbuf);
  // 2) 20000 columns, 16 per block
  trend_kernel<<<NTASK / 16, 256, 0, stream>>>(Qbuf, mQ);
  // 3) scalar reduce
  reduce_kernel<<<1, 256, 0, stream>>>(mQ, (float*)d_out);
}